// LazyCardiacDVFWarpModule_46617575031055
// MI455X (gfx1250) — compile-verified
//
#include <hip/hip_runtime.h>
#include <stdint.h>

// Problem geometry (fixed by the reference): D=H=W=160, N voxels = 4,096,000.
#define DX 160
#define HX 160
#define WX 160
#define NVOX (DX * HX * WX)
#define TILE 2048          // elements per TDM tile (per channel)
#define BLK 256            // 8 wave32 waves per block

typedef unsigned int v4u __attribute__((ext_vector_type(4)));
typedef int          v8i __attribute__((ext_vector_type(8)));
typedef int          v4i __attribute__((ext_vector_type(4)));

// ---------------------------------------------------------------------------
// TDM: DMA a 2D tile (TILE contiguous floats x 3 channel rows, row stride
// NVOX floats) from global memory into LDS at lds_byte_off. Tracked by
// TENSORcnt. Descriptor layout per CDNA5 ISA.
// ---------------------------------------------------------------------------
__device__ __forceinline__ void tdm_load_3rows(const float* gsrc, uint32_t lds_byte_off) {
  uint64_t ga = (uint64_t)(uintptr_t)gsrc;
  v4u g0;
  g0[0] = 1u;                                          // count=1, user descriptor
  g0[1] = lds_byte_off;                                // lds_addr [63:32]
  g0[2] = (uint32_t)(ga & 0xFFFFFFFFull);              // global_addr [95:64]
  g0[3] = (uint32_t)((ga >> 32) & 0x1FFFFFFull)        // global_addr [120:96]
        | (2u << 30);                                  // type=2 ("image") [127:126]
  v8i g1;
  g1[0] = (int)(2u << 16);                             // workgroup_mask=0, data_size=4B
  g1[1] = (int)((TILE & 0xFFFFu) << 16);               // tensor_dim0 lo16 at [63:48]
  g1[2] = (int)(((TILE >> 16) & 0xFFFFu)               // tensor_dim0 hi16
        | (3u << 16));                                 // tensor_dim1 = 3
  g1[3] = (int)((TILE & 0xFFFFu) << 16);               // tile_dim0 at [127:112]
  g1[4] = 3;                                           // tile_dim1=3, tile_dim2=0
  g1[5] = (int)(unsigned)NVOX;                         // tensor_dim0_stride lo32
  g1[6] = 0;                                           // stride0 hi16 | stride1 lo16
  g1[7] = 0;                                           // stride1 hi32
  v4i z4 = {0, 0, 0, 0};                               // groups 2/3 NULL (<=2D tile)
  v8i z8 = {0, 0, 0, 0, 0, 0, 0, 0};
  __builtin_amdgcn_tensor_load_to_lds(g0, g1, z4, z4, z8, 0);
}

// ---------------------------------------------------------------------------
// Trilinear helper pieces shared by the kernels.
// ---------------------------------------------------------------------------
struct TriSetup {
  int d0, h0, w0;
  float fd, fh, fw;
};

__device__ __forceinline__ TriSetup tri_setup(float cd, float ch, float cw) {
  TriSetup s;
  const float fd0 = floorf(cd), fh0 = floorf(ch), fw0 = floorf(cw);
  s.fd = cd - fd0; s.fh = ch - fh0; s.fw = cw - fw0;
  s.d0 = (int)fd0; s.h0 = (int)fh0; s.w0 = (int)fw0;
  return s;
}

// ---------------------------------------------------------------------------
// Kernel 1: blend dvf0/dvf1 (channel-planar) with scalar w, scale by 1/32,
// repack into float4-interleaved layout for B128 gathers downstream.
// TDM pulls both inputs' 3 channel rows into LDS asynchronously.
// ---------------------------------------------------------------------------
__global__ void __launch_bounds__(BLK)
blend_kernel(const float* __restrict__ dvf0, const float* __restrict__ dvf1,
             const float* __restrict__ wp, float4* __restrict__ out) {
  __shared__ float s0[3 * TILE];
  __shared__ float s1[3 * TILE];
  const int base = blockIdx.x * TILE;

  if (threadIdx.x < 32u) {                   // wave 0 issues the DMAs
    tdm_load_3rows(dvf0 + base, (uint32_t)(uintptr_t)s0);
    tdm_load_3rows(dvf1 + base, (uint32_t)(uintptr_t)s1);
    __builtin_amdgcn_s_wait_tensorcnt(0);    // s_wait_tensorcnt 0
  }
  __syncthreads();

  const float wv = wp[0];
  const float a = (1.0f - wv) * 0.03125f;    // fold the /2^5 into the blend
  const float b = wv * 0.03125f;
#pragma unroll 2
  for (int x = threadIdx.x; x < TILE; x += BLK) {
    float4 o;
    o.x = s0[x] * a + s1[x] * b;
    o.y = s0[TILE + x] * a + s1[TILE + x] * b;
    o.z = s0[2 * TILE + x] * a + s1[2 * TILE + x] * b;
    o.w = 0.0f;
    out[base + x] = o;
  }
}

// ---------------------------------------------------------------------------
// Kernel 2: one scaling-and-squaring step: dst = src + warp(src, src).
// 8-tap trilinear gather of a float4 field (one global_load_b128 per tap,
// L2-resident working set). Zeros padding via weight masking, clamp gather.
// ---------------------------------------------------------------------------
__global__ void __launch_bounds__(BLK)
step_kernel(const float4* __restrict__ src, float4* __restrict__ dst) {
  const int idx = blockIdx.x * BLK + threadIdx.x;   // grid covers NVOX exactly
  const int w = idx % WX;
  const int t = idx / WX;
  const int h = t % HX;
  const int d = t / HX;

  // Prefetch the stream ~256KB ahead (speculative global_prefetch_b8).
  {
    int pf = idx + 16384;
    pf = pf < NVOX ? pf : idx;
    __builtin_prefetch((const void*)(src + pf), 0, 0);
  }

  const float4 disp = src[idx];
  const TriSetup s = tri_setup((float)d + disp.x, (float)h + disp.y,
                               (float)w + disp.z);

  float ax = 0.0f, ay = 0.0f, az = 0.0f;
#pragma unroll
  for (int c = 0; c < 8; ++c) {
    const int dd = (c >> 2) & 1, dh = (c >> 1) & 1, dw = c & 1;
    const int di = s.d0 + dd, hi = s.h0 + dh, wi = s.w0 + dw;
    const bool inb = (di >= 0) & (di < DX) & (hi >= 0) & (hi < HX) &
                     (wi >= 0) & (wi < WX);
    const int dc = min(max(di, 0), DX - 1);
    const int hc = min(max(hi, 0), HX - 1);
    const int wc = min(max(wi, 0), WX - 1);
    float wgt = (dd ? s.fd : 1.0f - s.fd) * (dh ? s.fh : 1.0f - s.fh) *
                (dw ? s.fw : 1.0f - s.fw);
    wgt = inb ? wgt : 0.0f;
    const float4 v = src[(dc * HX + hc) * WX + wc];
    ax = fmaf(v.x, wgt, ax);
    ay = fmaf(v.y, wgt, ay);
    az = fmaf(v.z, wgt, az);
  }
  float4 o;
  o.x = disp.x + ax;
  o.y = disp.y + ay;
  o.z = disp.z + az;
  o.w = 0.0f;
  dst[idx] = o;
}

// ---------------------------------------------------------------------------
// Kernel 3 (fused): final squaring step + output warps. The finalize stage
// only reads the final ddf at its own voxel, so step 5 never needs to be
// materialized — compute disp_final in registers, then warp image (trilinear)
// and labels (nearest, rintf == jnp.round half-to-even) and write all outputs.
// Output stores are non-temporal: d_out is write-once and must not evict the
// L2-resident velocity field.
// out[0..N) img | out[N..2N) cavity | out[2N..3N) coronary | out[3N..6N) ddf planar
// ---------------------------------------------------------------------------
__global__ void __launch_bounds__(BLK)
step_final_kernel(const float4* __restrict__ src, const float* __restrict__ img,
                  const int* __restrict__ cav, const int* __restrict__ cor,
                  float* __restrict__ out) {
  const int idx = blockIdx.x * BLK + threadIdx.x;
  const int w = idx % WX;
  const int t = idx / WX;
  const int h = t % HX;
  const int d = t / HX;

  // Prefetch: field stream ahead, and the image neighborhood for this voxel
  // (displacements are small, so img[idx] is near the gather footprint).
  {
    int pf = idx + 16384;
    pf = pf < NVOX ? pf : idx;
    __builtin_prefetch((const void*)(src + pf), 0, 0);
    __builtin_prefetch((const void*)(img + idx), 0, 0);
  }

  // ---- step 5: disp_final = disp + warp(src)(voxel + disp) ----
  const float4 disp = src[idx];
  {
  }
  const TriSetup s5 = tri_setup((float)d + disp.x, (float)h + disp.y,
                                (float)w + disp.z);
  float ax = 0.0f, ay = 0.0f, az = 0.0f;
#pragma unroll
  for (int c = 0; c < 8; ++c) {
    const int dd = (c >> 2) & 1, dh = (c >> 1) & 1, dw = c & 1;
    const int di = s5.d0 + dd, hi = s5.h0 + dh, wi = s5.w0 + dw;
    const bool inb = (di >= 0) & (di < DX) & (hi >= 0) & (hi < HX) &
                     (wi >= 0) & (wi < WX);
    const int dc = min(max(di, 0), DX - 1);
    const int hc = min(max(hi, 0), HX - 1);
    const int wc = min(max(wi, 0), WX - 1);
    float wgt = (dd ? s5.fd : 1.0f - s5.fd) * (dh ? s5.fh : 1.0f - s5.fh) *
                (dw ? s5.fw : 1.0f - s5.fw);
    wgt = inb ? wgt : 0.0f;
    const float4 v = src[(dc * HX + hc) * WX + wc];
    ax = fmaf(v.x, wgt, ax);
    ay = fmaf(v.y, wgt, ay);
    az = fmaf(v.z, wgt, az);
  }
  const float fdx = disp.x + ax;          // final ddf at this voxel
  const float fdy = disp.y + ay;
  const float fdz = disp.z + az;

  const float cd = (float)d + fdx;
  const float ch = (float)h + fdy;
  const float cw = (float)w + fdz;

  // ---- trilinear image warp ----
  const TriSetup si = tri_setup(cd, ch, cw);
  float acc = 0.0f;
#pragma unroll
  for (int c = 0; c < 8; ++c) {
    const int dd = (c >> 2) & 1, dh = (c >> 1) & 1, dw = c & 1;
    const int di = si.d0 + dd, hi = si.h0 + dh, wi = si.w0 + dw;
    const bool inb = (di >= 0) & (di < DX) & (hi >= 0) & (hi < HX) &
                     (wi >= 0) & (wi < WX);
    const int dc = min(max(di, 0), DX - 1);
    const int hc = min(max(hi, 0), HX - 1);
    const int wc = min(max(wi, 0), WX - 1);
    float wgt = (dd ? si.fd : 1.0f - si.fd) * (dh ? si.fh : 1.0f - si.fh) *
                (dw ? si.fw : 1.0f - si.fw);
    wgt = inb ? wgt : 0.0f;
    acc = fmaf(img[(dc * HX + hc) * WX + wc], wgt, acc);
  }
  __builtin_nontemporal_store(acc, out + idx);

  // ---- nearest-neighbor label warps (round half to even == jnp.round) ----
  const int ni = (int)rintf(cd);
  const int nj = (int)rintf(ch);
  const int nk = (int)rintf(cw);
  const bool ninb = (ni >= 0) & (ni < DX) & (nj >= 0) & (nj < HX) &
                    (nk >= 0) & (nk < WX);
  const int nic = min(max(ni, 0), DX - 1);
  const int njc = min(max(nj, 0), HX - 1);
  const int nkc = min(max(nk, 0), WX - 1);
  const int nf = (nic * HX + njc) * WX + nkc;
  __builtin_nontemporal_store(ninb ? (float)cav[nf] : 0.0f, out + NVOX + idx);
  __builtin_nontemporal_store(ninb ? (float)cor[nf] : 0.0f, out + 2 * NVOX + idx);

  // ---- ddf output, channel planar ----
  __builtin_nontemporal_store(fdx, out + 3 * NVOX + idx);
  __builtin_nontemporal_store(fdy, out + 4 * NVOX + idx);
  __builtin_nontemporal_store(fdz, out + 5 * NVOX + idx);
}

// ---------------------------------------------------------------------------
extern "C" void kernel_launch(void* const* d_in, const int* in_sizes, int n_in,
                              void* d_out, int out_size, void* d_ws, size_t ws_size,
                              hipStream_t stream) {
  (void)in_sizes; (void)n_in; (void)out_size; (void)ws_size;
  const float* dvf0     = (const float*)d_in[0];
  const float* dvf1     = (const float*)d_in[1];
  const float* image    = (const float*)d_in[2];
  const int*   cavity   = (const int*)d_in[3];
  const int*   coronary = (const int*)d_in[4];
  const float* wp       = (const float*)d_in[5];
  float*       out      = (float*)d_out;

  float4* bufA = (float4*)d_ws;        // 65.5 MB
  float4* bufB = bufA + NVOX;          // 65.5 MB (both L2-resident: 131 MB < 192 MB)

  blend_kernel<<<NVOX / TILE, BLK, 0, stream>>>(dvf0, dvf1, wp, bufA);

  const int nb = NVOX / BLK;           // 16000 blocks, exact
  step_kernel<<<nb, BLK, 0, stream>>>(bufA, bufB);   // step 1
  step_kernel<<<nb, BLK, 0, stream>>>(bufB, bufA);   // step 2
  step_kernel<<<nb, BLK, 0, stream>>>(bufA, bufB);   // step 3
  step_kernel<<<nb, BLK, 0, stream>>>(bufB, bufA);   // step 4
  // step 5 fused with image/label warps + output packing (bufB never written)
  step_final_kernel<<<nb, BLK, 0, stream>>>(bufA, image, cavity, coronary, out);
}